// GraphusionVAE_66718021976790
// MI455X (gfx1250) — compile-verified
//
#include <hip/hip_runtime.h>
#include <hip/hip_bf16.h>

// ---------------------------------------------------------------------------
// GraphusionVAE forward for MI455X (gfx1250, wave32, WMMA).
//
// All matrix ops run through V_WMMA_F32_16X16X4_F32 (f32-exact, so numerics
// match the f32 reference). The model is memory-bound on MI455X (≈30 GFLOP
// vs a few hundred MB of HBM traffic at 23.3 TB/s), so f32 WMMA is the right
// precision; the big algorithmic win is factoring ed_o1 over the [B,N,N,512]
// pair tensor into P[n]+Q[m] (68.7 GFLOP + 536MB -> 2.4 GFLOP, fused).
// GEMM waves compute 16x64 tiles (4 accumulators) when Ncol allows, reusing
// the A fragment across 4 WMMAs to cut VMEM issue per WMMA.
//
// Input pointer order assumption:
//   d_in[0]=f_nodes [4,256,16], d_in[1]=f_edges [4,256,256,16],
//   d_in[2]=adj_mat (int), d_in[3]=eps [4,256,128],
//   d_in[4..] = params leaves in jax.tree_leaves order (dict keys sorted
//   recursively; layer dicts {'g','b','W','c'} flatten as W,b,c,g).
// ---------------------------------------------------------------------------

typedef __attribute__((ext_vector_type(2))) float v2f;
typedef __attribute__((ext_vector_type(8))) float v8f;

#define BC 4
#define NC 256
#define D_EDGE_C 16
#define D_HID_C 256
#define NHEADS_C 8
#define BN_C 1024L      /* B*N   */
#define BNN_C 262144L   /* B*N*N */

__device__ __forceinline__ v8f wmma_f32_4(v2f a, v2f b, v8f c) {
#if __has_builtin(__builtin_amdgcn_wmma_f32_16x16x4_f32)
  return __builtin_amdgcn_wmma_f32_16x16x4_f32(false, a, false, b, (short)0, c,
                                               false, false);
#else
#warning "wmma f32 16x16x4 builtin missing - scalar fallback (wrong results)"
  c[0] += a.x * b.x + a.y * b.y;
  return c;
#endif
}

struct GArgs {
  const float* A; const float* Bm; const float* bias; const float* Dadd; float* C;
  long a_rs, a_bo, a_bi;          // A: row stride (col stride 1), batch strides
  long b_rs, b_cs, b_bo, b_bi;    // B: row(K) stride, col stride, batch strides
  long d_rs, d_bo, d_bi;          // optional add tensor (col stride 1)
  long c_rs, c_bo, c_bi;          // C: row stride (col stride 1), batch strides
  int M, Ncol, K, ob, ib;         // dims; batch = ob*ib
  float alpha; int relu; int nt;  // nt = N-tiles per wave (1 or 4)
};

// Generic strided-batched GEMM: C = relu?( alpha*(A@B) + bias + D ).
// One wave per 16x(16*nt) output tile, K-loop in steps of 4 via WMMA
// f32 16x16x4; the A fragment is loaded once per K-step and reused across
// nt WMMAs (A-frag reuse, fewer VMEM issues per WMMA).
// A-frag: lane l: row l&15, K=kk..kk+1 with kk=(l>=16)*2.
// B-frag: lane l: col l&15, same K pair.  C/D: vgpr i -> row rbase+i, col l&15.
__global__ void gemm_wmma_k(GArgs g) {
  const int wv = threadIdx.x >> 5;
  const int lane = threadIdx.x & 31;
  const int tiles_m = g.M >> 4;
  const int ntg = (g.Ncol >> 4) / g.nt;   // N tile-groups
  const long tpb = (long)tiles_m * ntg;
  const long total = tpb * g.ob * g.ib;
  long t = (long)blockIdx.x * (blockDim.x >> 5) + wv;
  if (t >= total) return;                 // wave-uniform: EXEC stays all-ones
  const long bi = t / tpb;
  const long r = t - bi * tpb;
  const int tg = (int)(r % ntg);
  const int tm = (int)(r / ntg);
  const int ob = (int)(bi / g.ib);
  const int ib = (int)(bi - (long)ob * g.ib);
  const int tn0 = tg * g.nt;
  const int l15 = lane & 15;
  const int kk = (lane >> 4) << 1;
  const float* Arow = g.A + (long)ob * g.a_bo + (long)ib * g.a_bi +
                      ((long)tm * 16 + l15) * g.a_rs;
  const float* Bbase = g.Bm + (long)ob * g.b_bo + (long)ib * g.b_bi +
                       ((long)tn0 * 16 + l15) * g.b_cs;
  const int rbase = (lane < 16) ? 0 : 8;
  float* Cb = g.C + (long)ob * g.c_bo + (long)ib * g.c_bi;
  const float* Db = g.Dadd ? (g.Dadd + (long)ob * g.d_bo + (long)ib * g.d_bi)
                           : nullptr;

  if (g.nt == 4) {  // wide path: 16x64 per wave, A-frag reused 4x
    const float* B0 = Bbase;
    const float* B1 = Bbase + 16 * g.b_cs;
    const float* B2 = Bbase + 32 * g.b_cs;
    const float* B3 = Bbase + 48 * g.b_cs;
    v8f a0 = {}, a1 = {}, a2 = {}, a3 = {};
#pragma unroll 2
    for (int k = 0; k < g.K; k += 4) {
      const long ko0 = (long)(k + kk) * g.b_rs;
      const long ko1 = (long)(k + kk + 1) * g.b_rs;
      v2f a, b;
      a.x = Arow[k + kk];
      a.y = Arow[k + kk + 1];
      b.x = B0[ko0]; b.y = B0[ko1]; a0 = wmma_f32_4(a, b, a0);
      b.x = B1[ko0]; b.y = B1[ko1]; a1 = wmma_f32_4(a, b, a1);
      b.x = B2[ko0]; b.y = B2[ko1]; a2 = wmma_f32_4(a, b, a2);
      b.x = B3[ko0]; b.y = B3[ko1]; a3 = wmma_f32_4(a, b, a3);
    }
#pragma unroll
    for (int j = 0; j < 4; j++) {
      v8f acc = (j == 0) ? a0 : (j == 1) ? a1 : (j == 2) ? a2 : a3;
      const int nc = (tn0 + j) * 16 + l15;
      const float bv = g.bias ? g.bias[nc] : 0.f;
#pragma unroll
      for (int i = 0; i < 8; i++) {
        int row = tm * 16 + rbase + i;
        float v = acc[i] * g.alpha + bv;
        if (Db) v += Db[(long)row * g.d_rs + nc];
        if (g.relu) v = fmaxf(v, 0.f);
        Cb[(long)row * g.c_rs + nc] = v;
      }
    }
  } else {  // narrow path: 16x16 per wave
    v8f acc = {};
#pragma unroll 4
    for (int k = 0; k < g.K; k += 4) {
      v2f a, b;
      a.x = Arow[k + kk];
      a.y = Arow[k + kk + 1];
      b.x = Bbase[(long)(k + kk) * g.b_rs];
      b.y = Bbase[(long)(k + kk + 1) * g.b_rs];
      acc = wmma_f32_4(a, b, acc);
    }
    const int nc = tn0 * 16 + l15;
    const float bv = g.bias ? g.bias[nc] : 0.f;
#pragma unroll
    for (int i = 0; i < 8; i++) {
      int row = tm * 16 + rbase + i;
      float v = acc[i] * g.alpha + bv;
      if (Db) v += Db[(long)row * g.d_rs + nc];
      if (g.relu) v = fmaxf(v, 0.f);
      Cb[(long)row * g.c_rs + nc] = v;
    }
  }
}

// Y = [swish]( LN(X)*g + b [+ pos(row%N)] ), one wave per row.
__global__ void ln_act_k(const float* __restrict__ X, const float* __restrict__ g,
                         const float* __restrict__ b, const float* __restrict__ pos,
                         float* __restrict__ Y, int rows, int d, int do_swish) {
  int row = blockIdx.x * (blockDim.x >> 5) + (threadIdx.x >> 5);
  if (row >= rows) return;
  int lane = threadIdx.x & 31;
  const float* x = X + (long)row * d;
  float s = 0.f, s2 = 0.f;
  for (int j = lane; j < d; j += 32) { float v = x[j]; s += v; s2 += v * v; }
  for (int o = 16; o > 0; o >>= 1) { s += __shfl_xor(s, o, 32); s2 += __shfl_xor(s2, o, 32); }
  float mean = s / d;
  float var = s2 / d - mean * mean;
  float rstd = rsqrtf(var + 1e-5f);
  const float* pp = pos ? pos + (long)(row % NC) * d : nullptr;
  float* y = Y + (long)row * d;
  for (int j = lane; j < d; j += 32) {
    float v = (x[j] - mean) * rstd * g[j] + b[j];
    if (pp) v += pp[j];
    if (do_swish) v = v / (1.f + __expf(-v));
    y[j] = v;
  }
}

// Encoder masked softmax: score[b,h,n,m] = softmax_m( adj? feP[b,n,m,h] : -1e9 )
__global__ void enc_softmax_k(const float* __restrict__ feP,
                              const int* __restrict__ adj,
                              float* __restrict__ score) {
  int row = blockIdx.x * (blockDim.x >> 5) + (threadIdx.x >> 5);
  if (row >= BC * D_EDGE_C * NC) return;
  int lane = threadIdx.x & 31;
  int n = row % NC;
  int h = (row / NC) % D_EDGE_C;
  int b = row / (NC * D_EDGE_C);
  const int* am = adj + ((long)b * NC + n) * NC;
  const float* fp = feP + (((long)b * NC + n) * NC) * D_EDGE_C + h;
  float v[8], mx = -1e30f;
#pragma unroll
  for (int i = 0; i < 8; i++) {
    int m = lane + 32 * i;
    float t = (am[m] != 0) ? fp[(long)m * D_EDGE_C] : -1e9f;
    v[i] = t; mx = fmaxf(mx, t);
  }
  for (int o = 16; o > 0; o >>= 1) mx = fmaxf(mx, __shfl_xor(mx, o, 32));
  float s = 0.f;
#pragma unroll
  for (int i = 0; i < 8; i++) { v[i] = __expf(v[i] - mx); s += v[i]; }
  for (int o = 16; o > 0; o >>= 1) s += __shfl_xor(s, o, 32);
  float inv = 1.f / s;
  float* sc = score + ((long)((b * D_EDGE_C + h) * NC + n)) * NC;
#pragma unroll
  for (int i = 0; i < 8; i++) sc[lane + 32 * i] = v[i] * inv;
}

// Plain row softmax over 256 columns.
__global__ void softmax256_k(const float* __restrict__ X, float* __restrict__ Y,
                             int rows) {
  int row = blockIdx.x * (blockDim.x >> 5) + (threadIdx.x >> 5);
  if (row >= rows) return;
  int lane = threadIdx.x & 31;
  const float* x = X + (long)row * 256;
  float v[8], mx = -1e30f;
#pragma unroll
  for (int i = 0; i < 8; i++) { v[i] = x[lane + 32 * i]; mx = fmaxf(mx, v[i]); }
  for (int o = 16; o > 0; o >>= 1) mx = fmaxf(mx, __shfl_xor(mx, o, 32));
  float s = 0.f;
#pragma unroll
  for (int i = 0; i < 8; i++) { v[i] = __expf(v[i] - mx); s += v[i]; }
  for (int o = 16; o > 0; o >>= 1) s += __shfl_xor(s, o, 32);
  float inv = 1.f / s;
  float* y = Y + (long)row * 256;
#pragma unroll
  for (int i = 0; i < 8; i++) y[lane + 32 * i] = v[i] * inv;
}

__global__ void swish_k(float* X, long n) {
  long i = (long)blockIdx.x * blockDim.x + threadIdx.x;
  if (i < n) { float v = X[i]; X[i] = v / (1.f + __expf(-v)); }
}

__global__ void reparam_k(const float* __restrict__ mu, const float* __restrict__ lv,
                          const float* __restrict__ eps, float* __restrict__ z, long n) {
  long i = (long)blockIdx.x * blockDim.x + threadIdx.x;
  if (i < n) z[i] = mu[i] + eps[i] * __expf(0.5f * lv[i]);
}

// degrees = relu(x @ Wdeg + c), one wave per row.
__global__ void deg_k(const float* __restrict__ X, const float* __restrict__ W,
                      const float* __restrict__ c, float* __restrict__ out) {
  int row = blockIdx.x * (blockDim.x >> 5) + (threadIdx.x >> 5);
  if (row >= (int)BN_C) return;
  int lane = threadIdx.x & 31;
  const float* x = X + (long)row * D_HID_C;
  float s = 0.f;
  for (int j = lane; j < D_HID_C; j += 32) s += x[j] * W[j];
  for (int o = 16; o > 0; o >>= 1) s += __shfl_xor(s, o, 32);
  if (lane == 0) out[row] = fmaxf(s + c[0], 0.f);
}

// feo[b,n,m,:] = swish(P[b,n,:] + Q[b,m,:]) @ W2 + c2, fused (pair tensor
// never materialized). One wave per (b,n, m-tile of 16): A-frag generated
// on the fly, B = W2[256,16], K loop = 256.
__global__ void pair_feo_k(const float* __restrict__ P, const float* __restrict__ Q,
                           const float* __restrict__ W2, const float* __restrict__ c2,
                           float* __restrict__ out) {
  long t = (long)blockIdx.x * (blockDim.x >> 5) + (threadIdx.x >> 5);
  const long total = (long)BC * NC * (NC / 16);
  if (t >= total) return;
  int lane = threadIdx.x & 31;
  int tm = (int)(t % (NC / 16));
  long bn = t / (NC / 16);
  int b = (int)(bn / NC);
  int l15 = lane & 15;
  int kk = (lane >> 4) << 1;
  const float* Pp = P + bn * D_HID_C;
  const float* Qp = Q + ((long)b * NC + tm * 16 + l15) * D_HID_C;
  v8f acc = {};
#pragma unroll 4
  for (int k = 0; k < D_HID_C; k += 4) {
    float p0 = Pp[k + kk] + Qp[k + kk];
    float p1 = Pp[k + kk + 1] + Qp[k + kk + 1];
    v2f a, bv;
    a.x = p0 / (1.f + __expf(-p0));
    a.y = p1 / (1.f + __expf(-p1));
    bv.x = W2[(long)(k + kk) * D_EDGE_C + l15];
    bv.y = W2[(long)(k + kk + 1) * D_EDGE_C + l15];
    acc = wmma_f32_4(a, bv, acc);
  }
  int rbase = (lane < 16) ? 0 : 8;
  float bias = c2[l15];
#pragma unroll
  for (int i = 0; i < 8; i++) {
    int m = tm * 16 + rbase + i;
    out[((bn)*NC + m) * D_EDGE_C + l15] = acc[i] + bias;
  }
}

// ---------------------------------------------------------------------------
// Host side
// ---------------------------------------------------------------------------
static inline GArgs mkG(const float* A, long a_rs, const float* Bm, long b_rs,
                        long b_cs, const float* bias, float* C, long c_rs, int M,
                        int Ncol, int K) {
  GArgs g{};
  g.A = A; g.a_rs = a_rs; g.a_bo = 0; g.a_bi = 0;
  g.Bm = Bm; g.b_rs = b_rs; g.b_cs = b_cs; g.b_bo = 0; g.b_bi = 0;
  g.bias = bias; g.Dadd = nullptr; g.d_rs = 0; g.d_bo = 0; g.d_bi = 0;
  g.C = C; g.c_rs = c_rs; g.c_bo = 0; g.c_bi = 0;
  g.M = M; g.Ncol = Ncol; g.K = K; g.ob = 1; g.ib = 1;
  g.alpha = 1.f; g.relu = 0; g.nt = 1;
  return g;
}

static inline void gemm(hipStream_t s, GArgs g) {
  g.nt = (((g.Ncol >> 4) & 3) == 0) ? 4 : 1;
  long tiles = (long)(g.M >> 4) * ((g.Ncol >> 4) / g.nt) * g.ob * g.ib;
  long blocks = (tiles + 7) / 8;
  gemm_wmma_k<<<dim3((unsigned)blocks), dim3(256), 0, s>>>(g);
}

static inline void ln_act(hipStream_t s, const float* X, const float* g,
                          const float* b, const float* pos, float* Y, long rows,
                          int d, int sw) {
  long blocks = (rows + 7) / 8;
  ln_act_k<<<dim3((unsigned)blocks), dim3(256), 0, s>>>(X, g, b, pos, Y,
                                                        (int)rows, d, sw);
}

extern "C" void kernel_launch(void* const* d_in, const int* in_sizes, int n_in,
                              void* d_out, int out_size, void* d_ws, size_t ws_size,
                              hipStream_t stream) {
  (void)in_sizes; (void)n_in; (void)out_size; (void)ws_size;
  auto f = [&](int i) -> const float* { return (const float*)d_in[i]; };

  const float* f_nodes = f(0);
  const float* f_edges = f(1);
  const int* adjmat = (const int*)d_in[2];
  const float* eps = f(3);

  // ---- params leaf index table (jax.tree_leaves order, see header) ----
  // adj1.W=4 c=5 | adj2.W=6 c=7
  // dec[0]: b=8 ffW1=9 ffW2=10 ffb1=11 ffb2=12 ffc1=13 ffc2=14 ffg1=15 ffg2=16
  //         g=17 mW=18 mc=19 pos=20 qW=21 qc=22 scW=23 scc=24 scale=25
  // dec[k>=1]: 16 leaves starting 26+(k-1)*16 (same minus scW/scc)
  // deg.W=90 c=91 | ed_in.W=92 c=93 | ed_o1.W=94 c=95 | ed_o2.W=96 c=97
  // enc[k<4]: base 98+16k: edge_out{W,b,c,g} edge{..} node_out{..} node{..}
  // enc[4]: 162: edge{..} node_out{..} node{..}
  // logvar.W=174 c=175 | mu.W=176 c=177 | np1.W=178 c=179 | np2.W=180 c=181

  // ---- workspace layout (floats); needs ~103 MB ----
  float* ws = (float*)d_ws;
  long o = 0;
  auto alloc = [&](long nf) { float* p = ws + o; o += nf; return p; };
  float* FE0 = alloc(BNN_C * 16);   // edge ping (fe_cur / fe_next)
  float* FE1 = alloc(BNN_C * 16);   // edge proj (feP)
  float* BIG = alloc(BNN_C * 16);   // tmpE / enc score / dec score / a@aT
  float* NP  = alloc(BN_C * 2048);  // node_proj out [BN, 16*dn]
  float* AGG = alloc(BN_C * 2048);  // enc agg / dec attn-out
  float* TMP = alloc(BN_C * 2048);  // ln/swish scratch (node side)
  float* QKV = alloc(BN_C * 3072);
  float* T2  = alloc(BN_C * 512);
  float* T3  = alloc(BN_C * 512);
  float* HB  = alloc(BN_C * 256);
  float* X0  = alloc(BN_C * 256);
  float* X1  = alloc(BN_C * 256);
  float* SC  = alloc(BN_C * 256);
  float* PA  = alloc(BN_C * 256);
  float* QA  = alloc(BN_C * 256);
  float* FN0 = alloc(BN_C * 256);
  float* FN1 = alloc(BN_C * 256);
  float* TA  = alloc(BN_C * 256);
  float* TB  = alloc(BN_C * 256);

  float* outp = (float*)d_out;
  const long MU_OFF = 0, LV_OFF = 131072, FNO_OFF = 262144, FEO_OFF = 278528,
             ADJ_OFF = 4472832, DEG_OFF = 4734976;

  // ===================== Encoder =====================
  const int dn_tab[5] = {16, 16, 32, 64, 128};
  const int dno_tab[5] = {16, 32, 64, 128, 256};
  const float* fnCur = f_nodes;
  const float* feCur = f_edges;
  for (int k = 0; k < 5; k++) {
    const int dn = dn_tab[k], dno = dno_tab[k];
    const bool has_eo = (k < 4);
    const int base = (k < 4) ? (98 + 16 * k) : 162;
    // layers in block (W,b,c,g each): [edge_out], edge, node_out, node
    const int eoB = base, eB = has_eo ? base + 4 : base,
              noB = has_eo ? base + 8 : base + 4,
              nB = has_eo ? base + 12 : base + 8;
    // node_proj: TMP = swish(ln(fn)); NP = TMP @ W + c  [BN, 16*dn]
    ln_act(stream, fnCur, f(nB + 3), f(nB + 1), nullptr, TMP, BN_C, dn, 1);
    gemm(stream, mkG(TMP, dn, f(nB), 16L * dn, 1, f(nB + 2), NP, 16L * dn,
                     (int)BN_C, 16 * dn, dn));
    // edge_proj: BIG = swish(ln(fe)); FE1 = BIG @ W + c  [BNN,16]
    ln_act(stream, feCur, f(eB + 3), f(eB + 1), nullptr, BIG, BNN_C, 16, 1);
    gemm(stream, mkG(BIG, 16, f(eB), 16, 1, f(eB + 2), FE1, 16, (int)BNN_C, 16, 16));
    // masked softmax -> BIG [B,16,N,N]
    enc_softmax_k<<<dim3((BC * 16 * NC + 7) / 8), dim3(256), 0, stream>>>(
        FE1, adjmat, BIG);
    // agg[b,n,h*dn+d] = sum_m score[b,h,n,m] * NP[b,m,h*dn+d]
    {
      GArgs g = mkG(BIG, NC, NP, 16L * dn, 1, nullptr, AGG, 16L * dn, NC, dn, NC);
      g.ob = BC; g.ib = 16;
      g.a_bo = 16L * NC * NC; g.a_bi = (long)NC * NC;
      g.b_bo = (long)NC * 16 * dn; g.b_bi = dn;
      g.c_bo = (long)NC * 16 * dn; g.c_bi = dn;
      gemm(stream, g);
    }
    // node_out_proj: TMP = swish(ln(AGG)); fn_next = TMP @ W + c
    float* fnNext = (k & 1) ? FN1 : FN0;
    ln_act(stream, AGG, f(noB + 3), f(noB + 1), nullptr, TMP, BN_C, 16 * dn, 1);
    gemm(stream, mkG(TMP, 16L * dn, f(noB), dno, 1, f(noB + 2), fnNext, dno,
                     (int)BN_C, dno, 16 * dn));
    fnCur = fnNext;
    // edge_out_proj: BIG = swish(ln(FE1)); FE0 = BIG @ W + c
    if (has_eo) {
      ln_act(stream, FE1, f(eoB + 3), f(eoB + 1), nullptr, BIG, BNN_C, 16, 1);
      gemm(stream, mkG(BIG, 16, f(eoB), 16, 1, f(eoB + 2), FE0, 16, (int)BNN_C, 16, 16));
      feCur = FE0;
    }
  }

  // ===================== VAE head =====================
  gemm(stream, mkG(fnCur, 256, f(176), 128, 1, f(177), outp + MU_OFF, 128,
                   (int)BN_C, 128, 256));
  gemm(stream, mkG(fnCur, 256, f(174), 128, 1, f(175), outp + LV_OFF, 128,
                   (int)BN_C, 128, 256));
  reparam_k<<<dim3(512), dim3(256), 0, stream>>>(outp + MU_OFF, outp + LV_OFF,
                                                 eps, X0, BN_C * 128);

  // ===================== Decoder =====================
  for (int k = 0; k < 5; k++) {
    const int din = (k == 0) ? 128 : 256;
    const int dq = 2 * 512 + din * 8;  // 2048 or 3072
    const int bse = (k == 0) ? 8 : 26 + (k - 1) * 16;
    const float *Pb = f(bse + 0), *ffW1 = f(bse + 1), *ffW2 = f(bse + 2),
                *ffb1 = f(bse + 3), *ffb2 = f(bse + 4), *ffc1 = f(bse + 5),
                *ffc2 = f(bse + 6), *ffg1 = f(bse + 7), *ffg2 = f(bse + 8),
                *Pg = f(bse + 9), *mW = f(bse + 10), *mc = f(bse + 11),
                *pos = f(bse + 12), *qW = f(bse + 13), *qc = f(bse + 14);
    // h = swish(ln(x)+pos); qkv = h @ qkv_W + qkv_c
    ln_act(stream, X0, Pg, Pb, pos, HB, BN_C, din, 1);
    gemm(stream, mkG(HB, din, qW, dq, 1, qc, QKV, dq, (int)BN_C, dq, din));
    // score[b,h] = 0.125 * Q @ K^T  -> BIG [B,8,N,N]
    {
      GArgs g = mkG(QKV, dq, QKV + 512, 1, dq, nullptr, BIG, NC, NC, NC, 64);
      g.alpha = 0.125f;  // D_HEAD^-0.5
      g.ob = BC; g.ib = NHEADS_C;
      g.a_bo = (long)NC * dq; g.a_bi = 64;
      g.b_bo = (long)NC * dq; g.b_bi = 64;
      g.c_bo = (long)NHEADS_C * NC * NC; g.c_bi = (long)NC * NC;
      gemm(stream, g);
    }
    // out[b,n,h*din+c] = score @ V   -> AGG [BN, 8*din]
    {
      GArgs g = mkG(BIG, NC, QKV + 1024, dq, 1, nullptr, AGG,
                    (long)NHEADS_C * din, NC, din, NC);
      g.ob = BC; g.ib = NHEADS_C;
      g.a_bo = (long)NHEADS_C * NC * NC; g.a_bi = (long)NC * NC;
      g.b_bo = (long)NC * dq; g.b_bi = din;
      g.c_bo = (long)NC * NHEADS_C * din; g.c_bi = din;
      gemm(stream, g);
    }
    // x2 = x + out @ merge_W + merge_c  -> X1
    {
      GArgs g = mkG(AGG, (long)NHEADS_C * din, mW, din, 1, mc, X1, din,
                    (int)BN_C, din, NHEADS_C * din);
      g.Dadd = X0; g.d_rs = din;
      gemm(stream, g);
    }
    // ff: t2 = ln(x2)@W1+c1; t3 = swish(ln(t2)); x = t3@W2+c2 + shortcut
    ln_act(stream, X1, ffg1, ffb1, nullptr, HB, BN_C, din, 0);
    gemm(stream, mkG(HB, din, ffW1, 512, 1, ffc1, T2, 512, (int)BN_C, 512, din));
    ln_act(stream, T2, ffg2, ffb2, nullptr, T3, BN_C, 512, 1);
    const float* Dsc = X1;
    if (k == 0) {
      gemm(stream, mkG(X1, 128, f(23), 256, 1, f(24), SC, 256, (int)BN_C, 256, 128));
      Dsc = SC;
    }
    {
      GArgs g = mkG(T3, 512, ffW2, 256, 1, ffc2, X0, 256, (int)BN_C, 256, 512);
      g.Dadd = Dsc; g.d_rs = 256;
      gemm(stream, g);
    }
  }

  // ===================== Output heads =====================
  // degrees
  deg_k<<<dim3(128), dim3(256), 0, stream>>>(X0, f(90), f(91), outp + DEG_OFF);
  // adj = softmax( a @ a^T ),  a = adj2(swish(adj1(x)))
  gemm(stream, mkG(X0, 256, f(4), 256, 1, f(5), PA, 256, (int)BN_C, 256, 256));
  swish_k<<<dim3(1024), dim3(256), 0, stream>>>(PA, BN_C * 256);
  gemm(stream, mkG(PA, 256, f(6), 256, 1, f(7), QA, 256, (int)BN_C, 256, 256));
  {
    GArgs g = mkG(QA, 256, QA, 1, 256, nullptr, BIG, NC, NC, NC, 256);
    g.ob = BC; g.a_bo = (long)NC * 256; g.b_bo = (long)NC * 256;
    g.c_bo = (long)NC * NC;
    gemm(stream, g);
  }
  softmax256_k<<<dim3(128), dim3(256), 0, stream>>>(BIG, outp + ADJ_OFF, (int)BN_C);
  const float* adjO = outp + ADJ_OFF;
  // fno = np2( adj @ relu(np1(adj @ x)) )
  {
    GArgs g = mkG(adjO, NC, X0, 256, 1, nullptr, TA, 256, NC, 256, NC);
    g.ob = BC; g.a_bo = (long)NC * NC; g.b_bo = (long)NC * 256;
    g.c_bo = (long)NC * 256;
    gemm(stream, g);
  }
  {
    GArgs g = mkG(TA, 256, f(178), 256, 1, f(179), TB, 256, (int)BN_C, 256, 256);
    g.relu = 1;
    gemm(stream, g);
  }
  {
    GArgs g = mkG(adjO, NC, TB, 256, 1, nullptr, TA, 256, NC, 256, NC);
    g.ob = BC; g.a_bo = (long)NC * NC; g.b_bo = (long)NC * 256;
    g.c_bo = (long)NC * 256;
    gemm(stream, g);
  }
  gemm(stream, mkG(TA, 256, f(180), 16, 1, f(181), outp + FNO_OFF, 16,
                   (int)BN_C, 16, 256));
  // feo: h = ed_in(x); P = h@W1[0:256]+c1; Q = h@W1[256:512];
  // feo[b,n,m] = swish(P[b,n]+Q[b,m]) @ W2 + c2  (fused, pair never built)
  gemm(stream, mkG(X0, 256, f(92), 256, 1, f(93), HB, 256, (int)BN_C, 256, 256));
  gemm(stream, mkG(HB, 256, f(94), 256, 1, f(95), PA, 256, (int)BN_C, 256, 256));
  gemm(stream, mkG(HB, 256, f(94) + 256L * 256, 256, 1, nullptr, QA, 256,
                   (int)BN_C, 256, 256));
  pair_feo_k<<<dim3((unsigned)((BC * NC * (NC / 16) + 7) / 8)), dim3(256), 0,
               stream>>>(PA, QA, f(96), f(97), outp + FEO_OFF);
}